// TransformerConstrainedPooling_4363686773148
// MI455X (gfx1250) — compile-verified
//
#include <hip/hip_runtime.h>
#include <hip/hip_bf16.h>

// Problem constants (from the reference)
#define NROWS 131072
#define FDIM  128
#define HDIM  64
#define KCL   5
#define TCCOL 1280          // T*K = 256*5
#define HPAD  68            // h row pitch in LDS: 68*4 = 272B = 16B multiple, 68%64=4 -> conflict-free
#define WPITCH 80           // W1 pair-row pitch in float2: 160 dwords == 32 mod 64 banks
                            //  -> lanes 0-15 banks 0..31, lanes 16-31 banks 32..63 (conflict-free b64)

typedef __attribute__((ext_vector_type(2))) float v2f;
typedef __attribute__((ext_vector_type(4))) float f4;
typedef __attribute__((ext_vector_type(8))) float v8f;

__global__ __launch_bounds__(256) void tcp_mlp_scatter_kernel(
    const float* __restrict__ x,      // [N,128]
    const int*   __restrict__ ids,    // [N]
    const float* __restrict__ W1,     // [128,64]
    const float* __restrict__ b1,     // [64]
    const float* __restrict__ W2,     // [64,5]
    const float* __restrict__ b2,     // [5]
    float* __restrict__ S,            // [N,1280]
    float* __restrict__ cidBase,      // cluster ids region (after S)
    int cid64)                        // 1 -> write int64 ids, 0 -> int32
{
    // Aliased region:
    //   phase 1: W1 pairs, [64 pair-rows][WPITCH float2] = 40960 B
    //   phase 2+: h tiles, 8*16*HPAD floats = 34816 B
    __shared__ __align__(16) float smem[64 * WPITCH * 2];   // 40 KB
    __shared__ float b1s[HDIM];
    __shared__ float W2s[HDIM * KCL];
    __shared__ float b2s[KCL];
    __shared__ float Ps[8][16][KCL];
    __shared__ int   Col[8][16];

    const int tid  = threadIdx.x;
    const int w    = tid >> 5;        // wave id within block (0..7)
    const int lane = tid & 31;
    const int half = lane >> 4;       // 0: lanes 0-15, 1: lanes 16-31
    const int m    = lane & 15;

    // ---- Stage W1 into LDS as B-fragment-shaped float2 pairs ----
    // W1p[k2][col] = { W1[2*k2][col], W1[2*k2+1][col] }
    {
        v2f* W1p = (v2f*)smem;
        #pragma unroll
        for (int idx = tid; idx < 64 * 64; idx += 256) {
            const int k2  = idx >> 6;
            const int col = idx & 63;
            v2f pv;
            pv.x = W1[(2 * k2)     * HDIM + col];
            pv.y = W1[(2 * k2 + 1) * HDIM + col];
            W1p[k2 * WPITCH + col] = pv;
        }
    }
    if (tid < HDIM) b1s[tid] = b1[tid];
    for (int i = tid; i < HDIM * KCL; i += 256) W2s[i] = W2[i];
    if (tid < KCL) b2s[tid] = b2[tid];
    __syncthreads();

    const int rowbase = blockIdx.x * 128 + w * 16;   // this wave's 16 rows
    const float* xrow = x + (size_t)(rowbase + m) * FDIM;
    const v2f* W1p = (const v2f*)smem;

    // ---- Phase 1: h[16,64] = x_tile[16,128] @ W1 via V_WMMA_F32_16X16X4_F32 ----
    v8f acc[4];
    #pragma unroll
    for (int t = 0; t < 4; ++t)
        #pragma unroll
        for (int i = 0; i < 8; ++i) acc[t][i] = 0.0f;

    #pragma unroll 8
    for (int s = 0; s < 32; ++s) {
        const int kb = 4 * s;
        // A 16x4 fp32 fragment: lanes 0-15 hold K={kb,kb+1}, lanes 16-31 hold K={kb+2,kb+3}
        v2f a = *(const v2f*)(xrow + kb + 2 * half);
        // B fragment pair-row: k2 = kb/2 + half -> one ds_load_b64 per tile
        const v2f* brow = W1p + (2 * s + half) * WPITCH;
        #pragma unroll
        for (int t = 0; t < 4; ++t) {
            v2f b = brow[16 * t + m];
            acc[t] = __builtin_amdgcn_wmma_f32_16x16x4_f32(
                false, a, false, b, (short)0, acc[t], false, false);
        }
    }

    // ---- Stage h to LDS (re-using the W1 space), with bias + ReLU ----
    __syncthreads();   // everyone done reading W1 before the overwrite
    #pragma unroll
    for (int t = 0; t < 4; ++t) {
        const int col = 16 * t + m;
        const float bias = b1s[col];
        #pragma unroll
        for (int i = 0; i < 8; ++i) {
            const int r = i + 8 * half;                    // C/D layout: vgpr i holds row i + 8*half
            float v = acc[t][i] + bias;
            smem[(w * 16 + r) * HPAD + col] = v > 0.0f ? v : 0.0f;
        }
    }
    __syncthreads();

    // ---- Phase 2: logits = h @ W2 + b2, softmax, argmax (lanes 0-15, one row each) ----
    if (lane < 16) {
        const int r = lane;
        const float* hrow = &smem[(w * 16 + r) * HPAD];
        float lg[KCL];
        #pragma unroll
        for (int k = 0; k < KCL; ++k) lg[k] = b2s[k];
        #pragma unroll
        for (int c = 0; c < HDIM; c += 4) {
            f4 hv = *(const f4*)(hrow + c);
            #pragma unroll
            for (int e = 0; e < 4; ++e) {
                const float hval = hv[e];
                #pragma unroll
                for (int k = 0; k < KCL; ++k)
                    lg[k] = fmaf(hval, W2s[(c + e) * KCL + k], lg[k]);
            }
        }
        float mx = lg[0]; int kmax = 0;
        #pragma unroll
        for (int k = 1; k < KCL; ++k)
            if (lg[k] > mx) { mx = lg[k]; kmax = k; }   // strict > == first-occurrence argmax
        float es[KCL], sum = 0.0f;
        #pragma unroll
        for (int k = 0; k < KCL; ++k) { es[k] = __expf(lg[k] - mx); sum += es[k]; }
        const float inv = 1.0f / sum;
        const int id = ids[rowbase + r];
        const int cs = id * KCL;
        #pragma unroll
        for (int k = 0; k < KCL; ++k) Ps[w][r][k] = es[k] * inv;
        Col[w][r] = cs;
        const long long cid = (long long)(cs + kmax);
        if (cid64) ((long long*)cidBase)[rowbase + r] = cid;
        else       ((int*)cidBase)[rowbase + r]       = (int)cid;
    }
    __syncthreads();

    // ---- Phase 3: stream S rows (zeros + scattered P) with non-temporal b128 stores ----
    float* Sbase = S + (size_t)rowbase * TCCOL;
    const f4 z4 = {0.0f, 0.0f, 0.0f, 0.0f};
    #pragma unroll 1
    for (int r = 0; r < 16; ++r) {
        const int cs = Col[w][r];
        const float p0 = Ps[w][r][0], p1 = Ps[w][r][1], p2 = Ps[w][r][2],
                    p3 = Ps[w][r][3], p4 = Ps[w][r][4];
        const int lo = lane * 40;                       // this lane's 40-column slice
        float* rowp = Sbase + (size_t)r * TCCOL + lo;
        if (cs + 4 < lo || cs >= lo + 40) {
            #pragma unroll
            for (int j = 0; j < 10; ++j)
                __builtin_nontemporal_store(z4, (f4*)(rowp + 4 * j));
        } else {
            #pragma unroll
            for (int j = 0; j < 10; ++j) {
                f4 v;
                #pragma unroll
                for (int e = 0; e < 4; ++e) {
                    const int d = lo + 4 * j + e - cs;
                    float t = 0.0f;
                    t = (d == 0) ? p0 : t;
                    t = (d == 1) ? p1 : t;
                    t = (d == 2) ? p2 : t;
                    t = (d == 3) ? p3 : t;
                    t = (d == 4) ? p4 : t;
                    v[e] = t;
                }
                __builtin_nontemporal_store(v, (f4*)(rowp + 4 * j));
            }
        }
    }
}

extern "C" void kernel_launch(void* const* d_in, const int* in_sizes, int n_in,
                              void* d_out, int out_size, void* d_ws, size_t ws_size,
                              hipStream_t stream) {
    (void)in_sizes; (void)n_in; (void)d_ws; (void)ws_size;
    const float* x   = (const float*)d_in[0];
    const int*   ids = (const int*)d_in[1];
    const float* W1  = (const float*)d_in[2];
    const float* b1  = (const float*)d_in[3];
    const float* W2  = (const float*)d_in[4];
    const float* b2  = (const float*)d_in[5];

    float* S = (float*)d_out;
    float* cidBase = S + (size_t)NROWS * TCCOL;
    // Decide cluster_ids storage width from the tail of d_out.
    const long long rem = (long long)out_size - (long long)NROWS * TCCOL;
    const int cid64 = (rem >= 2LL * NROWS) ? 1 : 0;

    dim3 grid(NROWS / 128), block(256);
    hipLaunchKernelGGL(tcp_mlp_scatter_kernel, grid, block, 0, stream,
                       x, ids, W1, b1, W2, b2, S, cidBase, cid64);
}